// StateInferenceAgent_33681133535604
// MI455X (gfx1250) — compile-verified
//
#include <hip/hip_runtime.h>
#include <stdint.h>

// ---------------------------------------------------------------------------
// StateInferenceAgent forward scan for MI455X (gfx1250).
//
// 4096 independent sessions x 2048 sequential trials, 2-state belief each.
// Memory-bound (~167MB traffic ~= 7us at 23.3 TB/s) and latency-bound
// (parallelism capped at 4096 lanes = 128 wave32s), so each wave hides HBM
// latency itself with CDNA5 async global->LDS copies (ASYNCcnt), pipelined
// 2 chunks ahead across a triple-buffered LDS tile.
// ---------------------------------------------------------------------------

#define N_TRIALS 2048
#define TCHUNK   32                    // trials staged per chunk
#define NCHUNK   (N_TRIALS / TCHUNK)   // 64
#define ROWB     (TCHUNK * 12)         // 384 bytes of input per session/chunk
#define K16      (ROWB / 16)           // 24 x 16B pieces per session/chunk
#define NBUF     3                     // pipeline depth = NBUF-1 chunks ahead

typedef int v4i __attribute__((vector_size(4 * sizeof(int))));

// ---- CDNA5 async global->LDS copy (16B per lane) --------------------------
__device__ __forceinline__ void async_cp16(const void* g, void* l) {
#if __has_builtin(__builtin_amdgcn_global_load_async_to_lds_b128)
  __builtin_amdgcn_global_load_async_to_lds_b128((v4i*)g, (v4i*)l, 0, 0);
#else
  unsigned           lo = (unsigned)(uintptr_t)l;
  unsigned long long ga = (unsigned long long)(uintptr_t)g;
  asm volatile("global_load_async_to_lds_b128 %0, %1, off"
               :: "v"(lo), "v"(ga) : "memory");
#endif
}

#if __has_builtin(__builtin_amdgcn_s_wait_asynccnt)
#define WAIT_ASYNC(n) do { __builtin_amdgcn_s_wait_asynccnt(n); \
                           asm volatile("" ::: "memory"); } while (0)
#else
#define WAIT_ASYNC(n) asm volatile("s_wait_asynccnt %0" :: "i"(n) : "memory")
#endif

__global__ __launch_bounds__(32)
void StateInferenceAgent_scan_kernel(const float* __restrict__ inp,
                                     const float* __restrict__ p_stay_raw,
                                     const float* __restrict__ c_raw,
                                     float* __restrict__ out) {
  // LDS tile: [triple buffer][16B-slice k][lane] -> async writes are 32
  // consecutive 16B stores per instruction (bank-conflict free), compute
  // reads are lane-strided 16B (2-way conflict only).  36 KB total.
  __shared__ float4 lbuf[NBUF][K16][32];

  const int tid  = threadIdx.x;
  const int sess = blockIdx.x * 32 + tid;

  // Scalars (uniform): sigmoid of raw params, derived matrix entries.
  const float p  = 1.0f / (1.0f + __expf(-p_stay_raw[0]));
  const float c  = 1.0f / (1.0f + __expf(-c_raw[0]));
  const float ea = 0.5f * (1.0f + c);   // evidence "match" weight
  const float eb = 0.5f * (1.0f - c);   // evidence "mismatch" weight
  const float tp = 0.5f * (1.0f + p);   // transition diag
  const float tm = 0.5f * (1.0f - p);   // transition off-diag

  const char* gsrc = (const char*)inp + (size_t)sess * (N_TRIALS * 12);
  float*      orow = out + (size_t)sess * (N_TRIALS * 2);

  float s0 = 0.5f, s1 = 0.5f;

  // Issue one chunk of async copies: lane `tid` moves its own session's
  // 16B slice k into lbuf[buf][k][tid].
  auto issue = [&](int kc, int buf) {
#pragma unroll
    for (int k = 0; k < K16; ++k)
      async_cp16(gsrc + kc * ROWB + k * 16, &lbuf[buf][k][tid]);
  };

  // Fetch a float at byte `r` of this lane's staged row in buffer `buf`.
  auto ldf = [&](int buf, int r) -> float {
    return ((const float*)(&lbuf[buf][r >> 4][tid]))[(r & 15) >> 2];
  };

  // Prologue: prime the pipeline 2 chunks deep.
  issue(0, 0);
  issue(1, 1);

  int rbuf = 0;         // buffer holding chunk kc
  int wbuf = 2;         // buffer the next issue writes

  for (int kc = 0; kc < NCHUNK; ++kc) {
    if (kc + 2 < NCHUNK) {
      issue(kc + 2, wbuf);             // keep 2 chunks in flight
      wbuf = (wbuf == NBUF - 1) ? 0 : wbuf + 1;
      WAIT_ASYNC(2 * K16);             // in-order completion => chunk kc done
    } else if (kc + 1 < NCHUNK) {
      WAIT_ASYNC(K16);
    } else {
      WAIT_ASYNC(0);
    }

    float prev0 = 0.0f, prev1 = 0.0f;
#pragma unroll
    for (int t = 0; t < TCHUNK; ++t) {
      const int r = t * 12;
      const float cl = ldf(rbuf, r);        // chose_left  (exact 0/1)
      const float o  = ldf(rbuf, r + 8);    // outcome     (exact 0/1)
      // chose_right == 1 - cl exactly; evidence row collapses to a select:
      const bool  match = (cl == o);
      const float e0 = match ? ea : eb;
      const float e1 = match ? eb : ea;

      float l0 = e0 * s0, l1 = e1 * s1;
      float ls = l0 + l1;
      const bool z = (ls == 0.0f);
      l0 = z ? s0 : l0;
      l1 = z ? s1 : l1;
      ls = z ? 1.0f : ls;

      float inv = __builtin_amdgcn_rcpf(ls);
      inv = inv * __builtin_fmaf(-ls, inv, 2.0f);   // 1 NR step -> ~0.5 ulp
      const float n0 = l0 * inv;
      const float n1 = l1 * inv;

      s0 = __builtin_fmaf(n0, tp, n1 * tm);
      s1 = __builtin_fmaf(n0, tm, n1 * tp);

      if (t & 1) {                   // pack 2 trials -> one 16B store
        float4 v = make_float4(prev0, prev1, s0, s1);
        *(float4*)(orow + kc * (TCHUNK * 2) + (t - 1) * 2) = v;
      } else {
        prev0 = s0; prev1 = s1;
      }
    }
    rbuf = (rbuf == NBUF - 1) ? 0 : rbuf + 1;
  }
}

extern "C" void kernel_launch(void* const* d_in, const int* in_sizes, int n_in,
                              void* d_out, int out_size, void* d_ws, size_t ws_size,
                              hipStream_t stream) {
  const float* inp = (const float*)d_in[0];
  const float* pr  = (const float*)d_in[1];
  const float* cr  = (const float*)d_in[2];
  float*       out = (float*)d_out;

  const int n_sess = in_sizes[0] / (N_TRIALS * 3);   // 4096
  StateInferenceAgent_scan_kernel<<<dim3(n_sess / 32), dim3(32), 0, stream>>>(
      inp, pr, cr, out);
}